// Vector_Quantizer_69509750718555
// MI455X (gfx1250) — compile-verified
//
#include <hip/hip_runtime.h>
#include <hip/hip_bf16.h>

typedef __attribute__((ext_vector_type(2))) float v2f;
typedef __attribute__((ext_vector_type(4))) float v4f;
typedef __attribute__((ext_vector_type(8))) float v8f;

#define VQ_D      64
#define VQ_E      1024
#define VQ_B      256
#define VQ_K      512
#define VQ_NROWS  (VQ_B * VQ_K)      // 131072
#define COMMIT    0.25f

// ---------------------------------------------------------------------------
// Kernel 1: w2[e] = ||weight[e]||^2
// ---------------------------------------------------------------------------
__global__ __launch_bounds__(256) void vq_w2_kernel(const float* __restrict__ W,
                                                    float* __restrict__ w2) {
    int e = blockIdx.x * blockDim.x + threadIdx.x;
    if (e < VQ_E) {
        const float* row = W + (long)e * VQ_D;
        float s = 0.0f;
#pragma unroll 8
        for (int j = 0; j < VQ_D; ++j) { float w = row[j]; s += w * w; }
        w2[e] = s;
    }
}

// ---------------------------------------------------------------------------
// Kernel 2: per-row argmin over (||e||^2 - 2 x.e) using V_WMMA_F32_16X16X4_F32
// One wave32 per 16-row strip of X. 4 waves per block.
//
// A (16x4 f32) lane layout (ISA 7.12.2): lanes 0-15 hold M=lane, K={0,1};
// lanes 16-31 hold M=lane-16, K={2,3}. B (4x16) mirrors with N across lanes.
// C/D (16x16 f32): vgpr v, lanes 0-15 -> (M=v, N=lane); lanes 16-31 ->
// (M=v+8, N=lane-16).
//
// Two N-tiles in flight per loop iteration: the two WMMA accumulation chains
// (c0, c1) are independent, so the scheduler interleaves them and hides the
// WMMA->WMMA RAW latency instead of stalling on s_delay_alu VALU_DEP_1.
// ---------------------------------------------------------------------------
__global__ __launch_bounds__(128) void vq_argmin_wmma_kernel(
    const float* __restrict__ X, const float* __restrict__ W,
    const float* __restrict__ w2, int* __restrict__ Idx) {

    const int lane  = threadIdx.x & 31;
    const int wave  = threadIdx.x >> 5;
    const long strip = (long)blockIdx.x * 4 + wave;      // 16-row strip id
    const long rowBase = strip * 16;

    const int m    = lane & 15;         // A-row / B-col / C-col index for lane
    const int koff = (lane >> 4) * 2;   // K offset 0 or 2 per half-wave

    // Load the 16x64 A strip: 16 K-tiles of (K=4), 2 f32 per lane per tile.
    v2f a[16];
    const float* xrow = X + (rowBase + m) * VQ_D;
#pragma unroll
    for (int kt = 0; kt < 16; ++kt) {
        a[kt].x = xrow[kt * 4 + koff];
        a[kt].y = xrow[kt * 4 + koff + 1];
    }

    float best[8];
    int   bidx[8];
#pragma unroll
    for (int v = 0; v < 8; ++v) { best[v] = 3.4e38f; bidx[v] = 0; }

    for (int nt = 0; nt < VQ_E / 16; nt += 2) {
        const int nbase0 = nt * 16;
        const int nbase1 = nbase0 + 16;
        const float* wrow0 = W + (long)(nbase0 + m) * VQ_D;
        const float* wrow1 = W + (long)(nbase1 + m) * VQ_D;
        v8f c0 = {};
        v8f c1 = {};
#pragma unroll
        for (int kt = 0; kt < 16; ++kt) {
            v2f b0, b1;
            b0.x = wrow0[kt * 4 + koff];
            b0.y = wrow0[kt * 4 + koff + 1];
            b1.x = wrow1[kt * 4 + koff];
            b1.y = wrow1[kt * 4 + koff + 1];
            c0 = __builtin_amdgcn_wmma_f32_16x16x4_f32(
                     false, a[kt], false, b0, (short)0, c0, false, false);
            c1 = __builtin_amdgcn_wmma_f32_16x16x4_f32(
                     false, a[kt], false, b1, (short)0, c1, false, false);
        }
        const float wsq0 = w2[nbase0 + m];   // lane's N column, tile 0
        const float wsq1 = w2[nbase1 + m];   // lane's N column, tile 1
        const int   n0   = nbase0 + m;
        const int   n1   = nbase1 + m;
        // Tile nt first, then nt+1: strict '<' keeps earliest index (argmin).
#pragma unroll
        for (int v = 0; v < 8; ++v) {
            float s0 = __builtin_fmaf(-2.0f, c0[v], wsq0);
            if (s0 < best[v]) { best[v] = s0; bidx[v] = n0; }
            float s1 = __builtin_fmaf(-2.0f, c1[v], wsq1);
            if (s1 < best[v]) { best[v] = s1; bidx[v] = n1; }
        }
    }

    // Argmin reduction across the 16 lanes of each half-wave (N dimension).
    // Tie-break: smaller index wins (matches jnp.argmin).
#pragma unroll
    for (int v = 0; v < 8; ++v) {
        float s = best[v];
        int   n = bidx[v];
#pragma unroll
        for (int off = 8; off >= 1; off >>= 1) {
            float so = __shfl_xor(s, off, 32);
            int   no = __shfl_xor(n, off, 32);
            if (so < s || (so == s && no < n)) { s = so; n = no; }
        }
        if (lane == 0)  Idx[rowBase + v]     = n;   // rows M=0..7
        if (lane == 16) Idx[rowBase + 8 + v] = n;   // rows M=8..15
    }
}

// ---------------------------------------------------------------------------
// Kernel 3: quantized_st[row] = (masked) weight[idx[row]]; partial loss sums.
// 16 rows per 256-thread block; thread t -> row t>>4, float4 segment t&15.
// ---------------------------------------------------------------------------
__global__ __launch_bounds__(256) void vq_quantize_loss_kernel(
    const float* __restrict__ X, const float* __restrict__ W,
    const int* __restrict__ Idx, const int* __restrict__ nKeep,
    const int* __restrict__ ftPtr, float* __restrict__ qOut,
    float* __restrict__ partial) {

    __shared__ float sred[256];
    const int  t   = threadIdx.x;
    const long row = (long)blockIdx.x * 16 + (t >> 4);
    const int  seg = (t & 15) * 4;

    const int  idx = Idx[row];
    const int  b   = (int)(row >> 9);     // row / 512
    const int  k   = (int)(row & 511);    // row % 512
    const bool keep = (ftPtr[0] == 0) || (k < nKeep[b]);

    v4f w = *(const v4f*)(W + (long)idx * VQ_D + seg);
    v4f x = *(const v4f*)(X + row * VQ_D + seg);
    v4f q = keep ? w : (v4f){0.0f, 0.0f, 0.0f, 0.0f};

    // quantized_st = inputs + stop_grad(quantized - inputs) == quantized (value)
    *(v4f*)(qOut + row * VQ_D + seg) = q;

    v4f d = q - x;
    sred[t] = d.x * d.x + d.y * d.y + d.z * d.z + d.w * d.w;
    __syncthreads();
#pragma unroll
    for (int off = 128; off >= 1; off >>= 1) {
        if (t < off) sred[t] += sred[t + off];
        __syncthreads();
    }
    if (t == 0) partial[blockIdx.x] = sred[0];
}

// ---------------------------------------------------------------------------
// Kernel 4: deterministic final loss reduce. loss = 1.25 * mean((q-x)^2)
// (e_latent == q_latent numerically; stop_gradient doesn't change values)
// ---------------------------------------------------------------------------
__global__ __launch_bounds__(256) void vq_loss_final_kernel(
    const float* __restrict__ partial, int np, float* __restrict__ out) {
    __shared__ float sred[256];
    float s = 0.0f;
    for (int i = threadIdx.x; i < np; i += 256) s += partial[i];
    sred[threadIdx.x] = s;
    __syncthreads();
#pragma unroll
    for (int off = 128; off >= 1; off >>= 1) {
        if (threadIdx.x < off) sred[threadIdx.x] += sred[threadIdx.x + off];
        __syncthreads();
    }
    if (threadIdx.x == 0)
        out[0] = sred[0] * (1.0f + COMMIT) / (float)((long)VQ_NROWS * VQ_D);
}

// ---------------------------------------------------------------------------
// Kernel 5: one-hot encodings, 512 MB streaming write with nontemporal stores.
// One block per row; thread t writes cols [4t, 4t+4).
// ---------------------------------------------------------------------------
__global__ __launch_bounds__(256) void vq_encodings_kernel(
    const int* __restrict__ Idx, float* __restrict__ enc) {
    const long row = blockIdx.x;
    const int  idx = Idx[row];
    const int  c0  = threadIdx.x * 4;
    v4f v = {0.0f, 0.0f, 0.0f, 0.0f};
    if (idx >= c0 && idx < c0 + 4) v[idx - c0] = 1.0f;
    __builtin_nontemporal_store(v, (v4f*)(enc + row * (long)VQ_E + c0));
}

// ---------------------------------------------------------------------------
extern "C" void kernel_launch(void* const* d_in, const int* in_sizes, int n_in,
                              void* d_out, int out_size, void* d_ws, size_t ws_size,
                              hipStream_t stream) {
    const float* X     = (const float*)d_in[0];   // [256,512,64] f32
    const float* W     = (const float*)d_in[1];   // [1024,64] f32
    const int*   nKeep = (const int*)d_in[2];     // [256] i32
    const int*   ftPtr = (const int*)d_in[3];     // scalar i32

    float* out     = (float*)d_out;
    float* lossOut = out;                                   // 1 elem
    float* qOut    = out + 1;                               // 131072*64
    float* encOut  = out + 1 + (long)VQ_NROWS * VQ_D;       // 131072*1024

    char*  ws      = (char*)d_ws;
    int*   Idx     = (int*)ws;                              // 131072 ints
    float* w2      = (float*)(ws + (size_t)VQ_NROWS * sizeof(int)); // 1024 f
    float* partial = w2 + VQ_E;                             // 8192 f

    vq_w2_kernel<<<VQ_E / 256, 256, 0, stream>>>(W, w2);
    vq_argmin_wmma_kernel<<<VQ_NROWS / 16 / 4, 128, 0, stream>>>(X, W, w2, Idx);
    vq_quantize_loss_kernel<<<VQ_NROWS / 16, 256, 0, stream>>>(
        X, W, Idx, nKeep, ftPtr, qOut, partial);
    vq_loss_final_kernel<<<1, 256, 0, stream>>>(partial, VQ_NROWS / 16, lossOut);
    vq_encodings_kernel<<<VQ_NROWS, 256, 0, stream>>>(Idx, encOut);
}